// StructNConv2D_d_with_s_76270029242954
// MI455X (gfx1250) — compile-verified
//
#include <hip/hip_runtime.h>

// StructNConv2D_d_with_s fused forward for MI455X (gfx1250, wave32).
// Memory-bound (~570MB @ 23.3TB/s ~ 24us); fp32 throughout; channel-mix via
// V_WMMA_F32_16X16X4_F32 (Cin=32 as 8 K-steps), exact fp32 accumulation.
// Divisions use v_rcp_f32 (~1ulp) instead of the IEEE div expansion: the
// denominators are eps-guarded sums, and cutting ~300 VALU/lane leaves more
// issue slots for hiding the s_prod_roll HBM stream.

typedef float v2f __attribute__((ext_vector_type(2)));
typedef float v8f __attribute__((ext_vector_type(8)));

#define EPSV 1e-20f

constexpr int Bn = 4, Cin = 32, CoutN = 32, Hn = 256, Wn = 256, KK = 9;
constexpr int WAVES_PER_BLOCK = 4;
constexpr int PLANE = Hn * Wn;                 // 65536
constexpr int OUT_HALF = Bn * CoutN * PLANE;   // 8388608 floats per output tensor

__device__ __forceinline__ float softplus_f(float x) {
  return (x > 0.f) ? (x + log1pf(__expf(-x))) : log1pf(__expf(x));
}

__device__ __forceinline__ float fast_rcp(float x) {
  return __builtin_amdgcn_rcpf(x);   // v_rcp_f32, ~1 ulp
}

// ws[0..287]   = softplus(spatial_weight)   [Cin=32][9]
// ws[288..1311]= softplus(channel_weight)   [Cout=32][Cin=32]
// ws[1312]     = 1/sum(sw) ; ws[1313] = 1/sum(cw)  (STRIDE=1)
__global__ __launch_bounds__(1024) void precompute_weights(
    const float* __restrict__ sp_w, const float* __restrict__ ch_w,
    float* __restrict__ ws) {
  __shared__ float s1[1024];
  __shared__ float s2[1024];
  int t = threadIdx.x;
  float swv = 0.f;
  if (t < 288) { swv = softplus_f(sp_w[t]); ws[t] = swv; }
  float cwv = softplus_f(ch_w[t]);
  ws[288 + t] = cwv;
  s1[t] = swv; s2[t] = cwv;
  __syncthreads();
  for (int off = 512; off > 0; off >>= 1) {
    if (t < off) { s1[t] += s1[t + off]; s2[t] += s2[t + off]; }
    __syncthreads();
  }
  if (t == 0) { ws[1312] = 1.f / s1[0]; ws[1313] = 1.f / s2[0]; }
}

__global__ __launch_bounds__(128) void fused_structnconv(
    const float* __restrict__ d, const float* __restrict__ cd,
    const float* __restrict__ sprod, const float* __restrict__ bias,
    const float* __restrict__ ws, float* __restrict__ out) {
  __shared__ float sw_s[288];
  __shared__ float cw_s[1024];
  __shared__ float trans[WAVES_PER_BLOCK][16][17];  // padded vs bank conflicts

  const int tid = threadIdx.x;
  for (int i = tid; i < 288; i += 128)  sw_s[i] = ws[i];
  for (int i = tid; i < 1024; i += 128) cw_s[i] = ws[288 + i];
  __syncthreads();
  const float inv_sum_sw = ws[1312];
  const float inv_sum_cw = ws[1313];

  const int wave  = tid >> 5;
  const int lane  = tid & 31;
  const int m     = lane & 15;   // pixel-in-tile (WMMA M for A, N for C/D)
  const int khalf = lane >> 4;   // selects K pair {2*khalf, 2*khalf+1}

  const int tile = blockIdx.x * WAVES_PER_BLOCK + wave;  // B*H*(W/16) tiles
  const int wt = tile & 15;
  const int h  = (tile >> 4) & 255;
  const int b  = tile >> 12;
  const int w0 = wt * 16;
  const int w  = w0 + m;

  // ---- Stage 1: depthwise 3x3 normalized conv -> WMMA A operands ----
  // A-matrix f32 16x4 layout: lane(m,khalf): vgpr0 = K=2*khalf, vgpr1 = K=2*khalf+1
  v2f xa[8];  // cd_spatial * d_spatial, channels c = 4*kk + 2*khalf + {0,1}
  v2f ya[8];  // cd_spatial
  #pragma unroll
  for (int kk = 0; kk < 8; ++kk) {
    #pragma unroll
    for (int j = 0; j < 2; ++j) {
      const int c = kk * 4 + khalf * 2 + j;
      const float* swc = &sw_s[c * KK];
      const int rowBase = (b * Cin + c) * Hn;                        // in rows
      const int spBase  = ((b * Cin + c) * KK) * PLANE + h * Wn + w; // +k*PLANE
      float nom = 0.f, den = 0.f;
      #pragma unroll
      for (int dy = -1; dy <= 1; ++dy) {
        const int hh = h + dy;
        if ((unsigned)hh < (unsigned)Hn) {
          const int rb = (rowBase + hh) * Wn;
          #pragma unroll
          for (int dx = -1; dx <= 1; ++dx) {
            const int wwp = w + dx;
            if ((unsigned)wwp < (unsigned)Wn) {
              const int k   = (dy + 1) * 3 + (dx + 1);
              const float dv  = d[rb + wwp];
              const float cdv = cd[rb + wwp];
              const float sp  = __builtin_nontemporal_load(&sprod[spBase + k * PLANE]);
              const float cpw = cdv * sp * swc[k];
              nom = fmaf(cpw, dv, nom);
              den += cpw;
            }
          }
        }
      }
      const float dsp = nom * fast_rcp(den + EPSV);
      const float csp = den * inv_sum_sw;
      const float xv  = csp * dsp;
      if (j == 0) { xa[kk].x = xv; ya[kk].x = csp; }
      else        { xa[kk].y = xv; ya[kk].y = csp; }
    }
  }

  // ---- Stage 2: 1x1 channel mixing via V_WMMA_F32_16X16X4_F32 ----
  // D[pix(M), o(N)] = sum_c A[pix,c] * B[c,o], B[c,o] = softplus(cw)[o,c]
  #pragma unroll
  for (int nt = 0; nt < 2; ++nt) {
    const int o = nt * 16 + m;   // this lane's Cout column (N = lane&15)
    v2f bw[8];
    #pragma unroll
    for (int kk = 0; kk < 8; ++kk) {
      const int c = kk * 4 + khalf * 2;
      bw[kk].x = cw_s[o * Cin + c];
      bw[kk].y = cw_s[o * Cin + c + 1];
    }
    v8f accX = {};  // nom2
    v8f accY = {};  // denom2
    #pragma unroll
    for (int kk = 0; kk < 8; ++kk)
      accX = __builtin_amdgcn_wmma_f32_16x16x4_f32(false, xa[kk], false, bw[kk],
                                                   (short)0, accX, false, false);
    #pragma unroll
    for (int kk = 0; kk < 8; ++kk)
      accY = __builtin_amdgcn_wmma_f32_16x16x4_f32(false, ya[kk], false, bw[kk],
                                                   (short)0, accY, false, false);

    const float biasv = bias[o];

    // d_out = accX/(accY+eps)+bias ; LDS transpose for coalesced stores.
    // C/D layout: lane(n=m,khalf), vgpr r -> (pix = r+8*khalf, o = n+16*nt)
    #pragma unroll
    for (int r = 0; r < 8; ++r) {
      const int p = r + 8 * khalf;
      trans[wave][m][p] = fmaf(accX[r], fast_rcp(accY[r] + EPSV), biasv);
    }
    #pragma unroll
    for (int rr = 0; rr < 8; ++rr) {   // same-wave DS ops are in-order
      const int ol = rr * 2 + khalf;
      const float v = trans[wave][ol][m];
      __builtin_nontemporal_store(
          v, &out[((b * CoutN + nt * 16 + ol) * Hn + h) * Wn + w0 + m]);
    }

    // cd_out = accY / sum(cw)
    #pragma unroll
    for (int r = 0; r < 8; ++r) {
      const int p = r + 8 * khalf;
      trans[wave][m][p] = accY[r] * inv_sum_cw;
    }
    #pragma unroll
    for (int rr = 0; rr < 8; ++rr) {
      const int ol = rr * 2 + khalf;
      const float v = trans[wave][ol][m];
      __builtin_nontemporal_store(
          v, &out[OUT_HALF + ((b * CoutN + nt * 16 + ol) * Hn + h) * Wn + w0 + m]);
    }
  }
}

extern "C" void kernel_launch(void* const* d_in, const int* in_sizes, int n_in,
                              void* d_out, int out_size, void* d_ws, size_t ws_size,
                              hipStream_t stream) {
  const float* d     = (const float*)d_in[0];
  const float* cd    = (const float*)d_in[1];
  // d_in[2] (s), d_in[3] (cs) unused by the forward pass
  const float* sprod = (const float*)d_in[4];
  const float* chw   = (const float*)d_in[5];
  const float* spw   = (const float*)d_in[6];
  const float* bias  = (const float*)d_in[7];
  float* out = (float*)d_out;
  float* ws  = (float*)d_ws;

  precompute_weights<<<1, 1024, 0, stream>>>(spw, chw, ws);

  const int tiles = Bn * Hn * (Wn / 16);          // 16384 wave-tiles
  fused_structnconv<<<tiles / WAVES_PER_BLOCK, 128, 0, stream>>>(
      d, cd, sprod, bias, ws, out);
}